// NLSPN_27977416966372
// MI455X (gfx1250) — compile-verified
//
#include <hip/hip_runtime.h>
#include <math.h>

typedef __attribute__((ext_vector_type(2))) float v2f;
typedef __attribute__((ext_vector_type(8))) float v8f;

#define BB   4
#define CGC  8
#define HH   240
#define WW   1216
#define NPIX (BB * HH * WW)   // 1,167,360
#define KTOT 72               // 8 ch * 3 * 3 taps
#define ASTR 136              // im2col LDS row stride (floats): 2*ASTR % 64 == 16
#define WSTR 73               // weight LDS row stride (floats): conflict-free

// ---------------------------------------------------------------------------
// Kernel A: 3x3 conv (8 -> 24 ch) via V_WMMA_F32_16X16X4_F32 + offset/affinity
// Block: 256 threads (8 waves), one (b, y) row x 128 columns.
// LDS holds an im2col tile ailds[K][pixel] so the per-chunk A address is
// abase + kc*(4*ASTR) -> pure immediate-offset ds_loads in the WMMA loop.
// ---------------------------------------------------------------------------
__global__ __launch_bounds__(256) void nlspn_conv_offaff(
    const float* __restrict__ guidance,   // (B, 8, H, W)
    const float* __restrict__ conv_w,     // (24, 8, 3, 3) = (24, 72)
    const float* __restrict__ conv_b,     // (24,)
    const float* __restrict__ aff_scale,  // (1,)
    float* __restrict__ offset_out,       // (B, 18, H, W)
    float* __restrict__ aff_out)          // (B, 9, H, W)
{
    __shared__ float ailds[KTOT * ASTR];  // im2col: [k][pixel], pixel 0..127
    __shared__ float wlds[32 * WSTR];     // weights, channels padded to 32
    __shared__ float blds[32];            // bias, padded
    __shared__ float cout[8 * 16 * 32];   // conv results [pixel][chan(pad 32)]

    const int tid = threadIdx.x;
    const int tile_x0 = blockIdx.x * 128;
    const int y = blockIdx.y;
    const int b = blockIdx.z;

    // --- stage weights (B-matrix source, row = out-chan, col = K) ---
    for (int e = tid; e < 32 * KTOT; e += 256) {
        int c = e / KTOT, k = e % KTOT;
        wlds[c * WSTR + k] = (c < 24) ? conv_w[c * KTOT + k] : 0.0f;
    }
    if (tid < 32) blds[tid] = (tid < 24) ? conv_b[tid] : 0.0f;

    // --- build im2col tile: ailds[k][p] = guidance[b, ci, y+kh-1, x0+p+kw-1]
    for (int e = tid; e < KTOT * ASTR; e += 256) {
        const int k = e / ASTR;
        const int p = e % ASTR;
        const int ci = k / 9;
        const int rr = k % 9;
        const int kh = rr / 3;
        const int kw = rr % 3;
        const int gy = y + kh - 1;
        const int gx = tile_x0 + p + kw - 1;
        float v = 0.0f;
        if (p < 128 && gy >= 0 && gy < HH && gx >= 0 && gx < WW)
            v = guidance[(((size_t)b * CGC + ci) * HH + gy) * WW + gx];
        ailds[e] = v;
    }
    __syncthreads();

    const int wv   = tid >> 5;
    const int lane = tid & 31;
    const int half = lane >> 4;   // K half-wave split
    const int nl   = lane & 15;   // M (A-frag / pixel) or N (B-frag / channel)
    const int koff = half * 2;    // lane's K sub-offset within a chunk

    // Per-lane invariant bases; the unrolled loop adds only constants.
    const int abase  = koff * ASTR + wv * 16 + nl;  // A: + kc*4*ASTR (+ASTR)
    const int b0base = nl * WSTR + koff;            // B tile 0: + kc*4 (+1)
    const int b1base = (16 + nl) * WSTR + koff;     // B tile 1: + kc*4 (+1)

    v8f acc0 = {};                // channels 0..15
    v8f acc1 = {};                // channels 16..31 (only 16..23 used)

#pragma unroll
    for (int kc = 0; kc < 18; ++kc) {
        v2f a, b0, b1;
        a.x  = ailds[abase + kc * 4 * ASTR];
        a.y  = ailds[abase + kc * 4 * ASTR + ASTR];
        b0.x = wlds[b0base + kc * 4];
        b0.y = wlds[b0base + kc * 4 + 1];
        b1.x = wlds[b1base + kc * 4];
        b1.y = wlds[b1base + kc * 4 + 1];
        acc0 = __builtin_amdgcn_wmma_f32_16x16x4_f32(
            false, a, false, b0, (short)0, acc0, false, false);
        acc1 = __builtin_amdgcn_wmma_f32_16x16x4_f32(
            false, a, false, b1, (short)0, acc1, false, false);
    }

    // D layout: VGPR r, lanes 0-15 -> M=r, lanes 16-31 -> M=8+r, N=nl
    // Branchless spill into 32-channel-padded tile (chan 24..31 unused junk).
#pragma unroll
    for (int r = 0; r < 8; ++r) {
        const int M = half * 8 + r;
        cout[(wv * 16 + M) * 32 + nl]      = acc0[r] + blds[nl];
        cout[(wv * 16 + M) * 32 + 16 + nl] = acc1[r] + blds[16 + nl];
    }
    __syncthreads();

    // --- per-pixel postprocessing: tanh + L1-normalize affinities ---
    if (tid < 128) {
        const int x = tile_x0 + tid;
        if (x < WW) {
            const float* oa = &cout[tid * 32];
            const float inv = 1.0f / (aff_scale[0] + 1e-8f);
            float af[8];
            float ssum = 0.0f;
#pragma unroll
            for (int j = 0; j < 8; ++j) {
                af[j] = tanhf(oa[16 + j]) * inv;
                ssum += fabsf(af[j]);
            }
            const float s = fmaxf(ssum + 1e-4f, 1.0f);
            float asum = 0.0f;
#pragma unroll
            for (int j = 0; j < 8; ++j) { af[j] /= s; asum += af[j]; }
            const float aref = 1.0f - asum;

#pragma unroll
            for (int k = 0; k < 9; ++k) {
                float av, oy, ox;
                if (k < 4)       { av = af[k];     oy = oa[k];     ox = oa[8 + k]; }
                else if (k == 4) { av = aref;      oy = 0.0f;      ox = 0.0f; }
                else             { av = af[k - 1]; oy = oa[k - 1]; ox = oa[8 + k - 1]; }
                aff_out[(((size_t)b * 9 + k) * HH + y) * WW + x] = av;
                offset_out[(((size_t)b * 18 + 2 * k)     * HH + y) * WW + x] = oy;
                offset_out[(((size_t)b * 18 + 2 * k + 1) * HH + y) * WW + x] = ox;
            }
        }
    }
}

// ---------------------------------------------------------------------------
// P0 = feat_init * confidence
// ---------------------------------------------------------------------------
__global__ __launch_bounds__(256) void nlspn_init_p(
    const float* __restrict__ feat, const float* __restrict__ conf,
    float* __restrict__ P)
{
    const int i = blockIdx.x * 256 + threadIdx.x;
    if (i < NPIX) P[i] = feat[i] * conf[i];
}

// ---------------------------------------------------------------------------
// One propagation step: nf = sum_k aff_k * bilinear(P, y+gy_k+offy, x+gx_k+offx)
// ---------------------------------------------------------------------------
__global__ __launch_bounds__(256) void nlspn_prop_step(
    const float* __restrict__ Pin,      // (B, H, W) feat*conf
    const float* __restrict__ conf,     // (B, 1, H, W)
    const float* __restrict__ offset,   // (B, 18, H, W)
    const float* __restrict__ aff,      // (B, 9, H, W)
    float* __restrict__ feat_out,       // (B, 1, H, W) slot in feats
    float* __restrict__ Pout)           // next-step feat*conf
{
    const int i = blockIdx.x * 256 + threadIdx.x;
    if (i >= NPIX) return;
    const int x = i % WW;
    const int t = i / WW;
    const int y = t % HH;
    const int b = t / HH;
    const float* Pb = Pin + (size_t)b * HH * WW;

    float nf = 0.0f;
#pragma unroll
    for (int k = 0; k < 9; ++k) {
        const float offy = offset[(((size_t)b * 18 + 2 * k)     * HH + y) * WW + x];
        const float offx = offset[(((size_t)b * 18 + 2 * k + 1) * HH + y) * WW + x];
        const float yy = (float)y + (float)(k / 3 - 1) + offy;
        const float xx = (float)x + (float)(k % 3 - 1) + offx;
        const float y0 = floorf(yy), x0 = floorf(xx);
        const float ty = yy - y0, tx = xx - x0;

        auto g = [&](float yf, float xf) -> float {
            if (yf < 0.0f || yf > (float)(HH - 1) ||
                xf < 0.0f || xf > (float)(WW - 1)) return 0.0f;
            return Pb[(int)yf * WW + (int)xf];
        };
        const float smp = (1.0f - ty) * (1.0f - tx) * g(y0,        x0)
                        + (1.0f - ty) * tx          * g(y0,        x0 + 1.0f)
                        + ty          * (1.0f - tx) * g(y0 + 1.0f, x0)
                        + ty          * tx          * g(y0 + 1.0f, x0 + 1.0f);
        const float a = aff[(((size_t)b * 9 + k) * HH + y) * WW + x];
        nf += a * smp;
    }
    feat_out[i] = nf;
    Pout[i] = nf * conf[i];
}

// ---------------------------------------------------------------------------
extern "C" void kernel_launch(void* const* d_in, const int* in_sizes, int n_in,
                              void* d_out, int out_size, void* d_ws, size_t ws_size,
                              hipStream_t stream)
{
    const float* feat_init  = (const float*)d_in[0];
    const float* guidance   = (const float*)d_in[1];
    const float* confidence = (const float*)d_in[2];
    const float* conv_w     = (const float*)d_in[3];
    const float* conv_b     = (const float*)d_in[4];
    const float* aff_scale  = (const float*)d_in[5];

    float* out = (float*)d_out;
    const size_t N = (size_t)NPIX;
    float* feat_final = out;            // N
    float* feats      = out + N;        // 18*N
    float* offset_out = out + 19 * N;   // 18*N
    float* aff_out    = out + 37 * N;   // 9*N
    float* scale_out  = out + 46 * N;   // 1

    float* P0 = (float*)d_ws;           // B*H*W f32
    float* P1 = P0 + N;

    // conv + offset/affinity (WMMA)
    nlspn_conv_offaff<<<dim3((WW + 127) / 128, HH, BB), 256, 0, stream>>>(
        guidance, conv_w, conv_b, aff_scale, offset_out, aff_out);

    // P0 = feat_init * confidence
    nlspn_init_p<<<(NPIX + 255) / 256, 256, 0, stream>>>(feat_init, confidence, P0);

    // 18 serial propagation steps (double-buffered P)
    for (int t = 0; t < 18; ++t) {
        const float* Pin = (t & 1) ? P1 : P0;
        float* Pout      = (t & 1) ? P0 : P1;
        nlspn_prop_step<<<NPIX / 256, 256, 0, stream>>>(
            Pin, confidence, offset_out, aff_out, feats + (size_t)t * N, Pout);
    }

    // feat_final = feats[17]; aff_scale pass-through
    hipMemcpyAsync(feat_final, feats + 17 * N, N * sizeof(float),
                   hipMemcpyDeviceToDevice, stream);
    hipMemcpyAsync(scale_out, aff_scale, sizeof(float),
                   hipMemcpyDeviceToDevice, stream);
}